// Group_17179869184137
// MI455X (gfx1250) — compile-verified
//
#include <hip/hip_runtime.h>

#define NB 16
#define NP 8192
#define NG 512
#define NK 32

typedef __attribute__((ext_vector_type(2))) float v2f;
typedef __attribute__((ext_vector_type(8))) float v8f;

// Output layout: neighborhood [B,G,K,3], center [B,G,3], gt_neighborhood [B,G,K,3], labels [B,G,K,1]
#define OFF_NBH ((size_t)0)
#define OFF_CEN ((size_t)NB * NG * NK * 3)
#define OFF_GTN (OFF_CEN + (size_t)NB * NG * 3)
#define OFF_LAB (OFF_GTN + (size_t)NB * NG * NK * 3)

// ---------------------------------------------------------------------------
// Kernel 1: farthest point sampling. One workgroup per batch.
//  - Whole 96KB point cloud async-copied into LDS once (ASYNCcnt path), so the
//    per-step broadcast read of the last center is an LDS hit on the serial
//    dependency chain instead of an L2 round trip.
//  - Each thread keeps its 8 points' coords and running min-distances entirely
//    in VGPRs -> zero per-step LDS dist traffic.
//  - Argmax: wave32 shfl_xor butterfly + 32-entry LDS stage (2 barriers/step).
// ---------------------------------------------------------------------------
__global__ __launch_bounds__(1024) void fps_kernel(const float* __restrict__ xyz,
                                                   int* __restrict__ fidx,
                                                   float* __restrict__ out) {
  const int b = blockIdx.x;
  const int tid = threadIdx.x;
  const int lane = tid & 31;
  const int wv = tid >> 5;
  const float* __restrict__ px = xyz + (size_t)b * NP * 3;

  __shared__ float ptsS[NP * 3];   // 96 KB point cloud (of 320KB/WGP)
  __shared__ float redD[32];
  __shared__ int   redI[32];
  __shared__ int   fS[NG];
  __shared__ int   lastS;

  // Async-stage the batch point cloud into LDS: 16B per lane per issue,
  // 6 issues per thread. Tracked by ASYNCcnt, drained before the barrier.
  for (int i = tid * 4; i < NP * 3; i += 1024 * 4) {
    const unsigned lds = (unsigned)(size_t)&ptsS[i];
    const unsigned long long ga = (unsigned long long)(px + i);
    asm volatile("global_load_async_to_lds_b128 %0, %1, off"
                 :: "v"(lds), "v"(ga) : "memory");
  }
  if (tid == 0) { lastS = 0; fS[0] = 0; }
  asm volatile("s_wait_asynccnt 0x0" ::: "memory");
  __syncthreads();

  // Thread-private coords + running min distances (registers only).
  float pr[8][3], dr[8];
#pragma unroll
  for (int j = 0; j < 8; ++j) {
    const int i = tid + 1024 * j;
    pr[j][0] = ptsS[3 * i + 0];
    pr[j][1] = ptsS[3 * i + 1];
    pr[j][2] = ptsS[3 * i + 2];
    dr[j] = 3.4e38f;
  }

  for (int it = 1; it < NG; ++it) {
    const int last = lastS;
    const float cx = ptsS[3 * last + 0];
    const float cy = ptsS[3 * last + 1];
    const float cz = ptsS[3 * last + 2];

    float bD = -1.0f;
    int   bI = 0;
#pragma unroll
    for (int j = 0; j < 8; ++j) {
      const float dx = pr[j][0] - cx;
      const float dy = pr[j][1] - cy;
      const float dz = pr[j][2] - cz;
      const float d = fminf(dr[j], dx * dx + dy * dy + dz * dz);
      dr[j] = d;
      if (d > bD) { bD = d; bI = tid + 1024 * j; }  // ascending idx + strict >
    }
    // wave32 butterfly argmax (ties -> lower index, matches jnp.argmax)
#pragma unroll
    for (int off = 16; off > 0; off >>= 1) {
      const float oD = __shfl_xor(bD, off, 32);
      const int   oI = __shfl_xor(bI, off, 32);
      if (oD > bD || (oD == bD && oI < bI)) { bD = oD; bI = oI; }
    }
    if (lane == 0) { redD[wv] = bD; redI[wv] = bI; }
    __syncthreads();
    if (wv == 0) {
      float rD = redD[lane];
      int   rI = redI[lane];
#pragma unroll
      for (int off = 16; off > 0; off >>= 1) {
        const float oD = __shfl_xor(rD, off, 32);
        const int   oI = __shfl_xor(rI, off, 32);
        if (oD > rD || (oD == rD && oI < rI)) { rD = oD; rI = oI; }
      }
      if (lane == 0) { lastS = rI; fS[it] = rI; }
    }
    __syncthreads();
  }

  // Emit fidx (workspace) and the center output (coords from LDS).
  for (int g = tid; g < NG; g += 1024) {
    const int id = fS[g];
    fidx[b * NG + g] = id;
    const size_t ob = OFF_CEN + ((size_t)b * NG + g) * 3;
    out[ob + 0] = ptsS[3 * id + 0];
    out[ob + 1] = ptsS[3 * id + 1];
    out[ob + 2] = ptsS[3 * id + 2];
  }
}

// ---------------------------------------------------------------------------
// Kernel 2: KNN via GEMM-form d2 = c^2 + p^2 - 2*c.p with the cross term on
// V_WMMA_F32_16X16X4_F32 (K=4: xyz padded with 0). One block = 16 centers of
// one batch, 8 waves; each wave sweeps 64 point-tiles of 16. All 32 lanes do
// selection: lane -> (center lane%16, column-half lane/16) replace-max top-32
// lists in LDS; final wave-parallel lexicographic-argmin merge.
// ---------------------------------------------------------------------------
__global__ __launch_bounds__(256) void knn_group_kernel(const float* __restrict__ xyz,
                                                        const float* __restrict__ gt,
                                                        const int* __restrict__ labels,
                                                        const int* __restrict__ fidx,
                                                        float* __restrict__ out) {
  const int b    = blockIdx.y;
  const int g0   = blockIdx.x * 16;
  const int tid  = threadIdx.x;
  const int w    = tid >> 5;     // wave 0..7
  const int lane = tid & 31;
  const int mm   = lane & 15;    // row/col within 16
  const int hh   = lane >> 4;    // half-wave
  const int k0   = hh * 2;       // K pair base for f32 16x4 / 4x16 fragments

  const float* __restrict__ px = xyz + (size_t)b * NP * 3;
  const float* __restrict__ pg = gt  + (size_t)b * NP * 3;

  __shared__ float cS[16][4];                // centers (K padded to 4 with 0)
  __shared__ float gS[16][3];                // gt centers
  __shared__ float c2S[16];                  // |c|^2
  __shared__ float d2S[8][16][16];           // per-wave d2 tile (8 KB)
  __shared__ float listD[8][2][16][NK];      // partial top-32 dists (32 KB)
  __shared__ int   listI[8][2][16][NK];      // partial top-32 indices (32 KB)
  __shared__ int   selI[16][NK];             // merged indices

  if (tid < 16) {
    const int id = fidx[b * NG + g0 + tid];
    const float x = px[3 * id + 0];
    const float y = px[3 * id + 1];
    const float z = px[3 * id + 2];
    cS[tid][0] = x; cS[tid][1] = y; cS[tid][2] = z; cS[tid][3] = 0.0f;
    c2S[tid] = x * x + y * y + z * z;
    gS[tid][0] = pg[3 * id + 0];
    gS[tid][1] = pg[3 * id + 1];
    gS[tid][2] = pg[3 * id + 2];
  }
  __syncthreads();

  // A fragment: 16x4 centers, lane -> row mm, VGPR pair -> K {k0, k0+1}.
  v2f a;
  a.x = cS[mm][k0];
  a.y = cS[mm][k0 + 1];

  float curMax = -1.0f;
  int curMaxPos = 0;
  int cnt = 0;

  for (int t = 0; t < 64; ++t) {
    const int n0 = (w * 64 + t) * 16;
    const int n = n0 + mm;
    const float* pp = px + 3 * (size_t)n;
    __builtin_prefetch(px + 3 * (size_t)(n + 16), 0, 0);  // global_prefetch_b8

    // B fragment: 4x16 points, lane -> col mm, VGPR pair -> K {k0,k0+1}, pad K=3 with 0.
    v2f bf;
    if (k0 == 0) { bf.x = pp[0]; bf.y = pp[1]; }
    else         { bf.x = pp[2]; bf.y = 0.0f;  }

    // |p|^2 per column via two wave shuffles (halves hold complementary K pairs).
    const float part = bf.x * bf.x + bf.y * bf.y;
    const float p2n = __shfl(part, mm, 32) + __shfl(part, mm + 16, 32);

    v8f acc = {};
    acc = __builtin_amdgcn_wmma_f32_16x16x4_f32(false, a, false, bf, (short)0, acc,
                                                false, false);

    // C layout: acc[v] at lane -> M = v + 8*hh, N = mm.
#pragma unroll
    for (int v = 0; v < 8; ++v) {
      const int m = v + 8 * hh;
      d2S[w][m][mm] = c2S[m] + p2n - 2.0f * acc[v];
    }
    __syncthreads();

    // All 32 lanes: lane owns (center mm, 8-column half hh); replace-max top-32.
    {
      const int m = mm;
      for (int q = 0; q < 8; ++q) {
        const int nn = 8 * hh + q;
        const float d = d2S[w][m][nn];
        const int gi = n0 + nn;
        if (cnt < NK) {
          listD[w][hh][m][cnt] = d;
          listI[w][hh][m][cnt] = gi;
          if (d > curMax) { curMax = d; curMaxPos = cnt; }
          ++cnt;
        } else if (d < curMax) {
          listD[w][hh][m][curMaxPos] = d;
          listI[w][hh][m][curMaxPos] = gi;
          float nmx = listD[w][hh][m][0];
          int nmp = 0;
          for (int j = 1; j < NK; ++j) {
            const float vv = listD[w][hh][m][j];
            if (vv > nmx) { nmx = vv; nmp = j; }
          }
          curMax = nmx;
          curMaxPos = nmp;
        }
      }
    }
    __syncthreads();
  }

  // Wave-parallel merge: wave w owns centers m = w and m = w+8. Per pick, 32
  // lanes scan 16 of the 512 candidates each, then a 5-step shfl_xor
  // lexicographic (d, idx) argmin — matches top_k ascending + lower-index ties.
  for (int cpair = 0; cpair < 2; ++cpair) {
    const int m = w + 8 * cpair;
    for (int j = 0; j < NK; ++j) {
      float bd = 3.4e38f;
      int bi = 0x7fffffff;
      int bc = lane;
      for (int c = lane; c < 512; c += 32) {
        const float d = listD[c >> 6][(c >> 5) & 1][m][c & 31];
        const int gi = listI[c >> 6][(c >> 5) & 1][m][c & 31];
        if (d < bd || (d == bd && gi < bi)) { bd = d; bi = gi; bc = c; }
      }
#pragma unroll
      for (int off = 16; off > 0; off >>= 1) {
        const float od = __shfl_xor(bd, off, 32);
        const int oi = __shfl_xor(bi, off, 32);
        const int oc = __shfl_xor(bc, off, 32);
        if (od < bd || (od == bd && oi < bi)) { bd = od; bi = oi; bc = oc; }
      }
      if (lane == 0) {
        selI[m][j] = bi;
        listD[bc >> 6][(bc >> 5) & 1][m][bc & 31] = 3.4e38f;  // consume winner
      }
      __builtin_amdgcn_wave_barrier();
    }
  }
  __syncthreads();

  // Gather + center-subtract + labels.
  for (int e = tid; e < 16 * NK; e += 256) {
    const int m = e >> 5;
    const int j = e & 31;
    const int gi = selI[m][j];
    const size_t gg = (size_t)b * NG + (g0 + m);
    const size_t ob = (gg * NK + j) * 3;
    out[OFF_NBH + ob + 0] = px[3 * gi + 0] - cS[m][0];
    out[OFF_NBH + ob + 1] = px[3 * gi + 1] - cS[m][1];
    out[OFF_NBH + ob + 2] = px[3 * gi + 2] - cS[m][2];
    out[OFF_GTN + ob + 0] = pg[3 * gi + 0] - gS[m][0];
    out[OFF_GTN + ob + 1] = pg[3 * gi + 1] - gS[m][1];
    out[OFF_GTN + ob + 2] = pg[3 * gi + 2] - gS[m][2];
    out[OFF_LAB + gg * NK + j] = (float)labels[(size_t)b * NP + gi];
  }
}

extern "C" void kernel_launch(void* const* d_in, const int* in_sizes, int n_in,
                              void* d_out, int out_size, void* d_ws, size_t ws_size,
                              hipStream_t stream) {
  const float* xyz    = (const float*)d_in[0];
  const float* gt     = (const float*)d_in[1];
  const int*   labels = (const int*)d_in[2];
  float* out = (float*)d_out;
  int*   fidx = (int*)d_ws;  // B*G int32 = 32 KB scratch

  fps_kernel<<<dim3(NB), dim3(1024), 0, stream>>>(xyz, fidx, out);
  knn_group_kernel<<<dim3(NG / 16, NB), dim3(256), 0, stream>>>(xyz, gt, labels, fidx, out);
}